// MLP_74586402063282
// MI455X (gfx1250) — compile-verified
//
#include <hip/hip_runtime.h>
#include <math.h>

typedef __attribute__((ext_vector_type(2))) float v2f;
typedef __attribute__((ext_vector_type(8))) float v8f;

#define KNB 16        // neighbors
#define HDIM 128      // hidden width
#define KPAD 136      // padded K for wmma chunks of 4
#define DOFF 8        // diff features start at col 8 (cols 5..7 zero)
#define S2STRIDE 140  // LDS row stride: even (aligned v2f/float4), 12*r%64 distinct
#define NPAIR (KPAD / 2)  // 68 packed row-pairs in w2pp

// w2 logical row for padded layout: 0..4 -> w2 rows 0..4, 5..7 -> zero,
// 8..135 -> w2 rows 5..132
__device__ __forceinline__ float w2row(const float* w2, int lr, int col) {
  if (lr < 5) return w2[lr * HDIM + col];
  if (lr < DOFF) return 0.0f;
  return w2[(lr - 3) * HDIM + col];
}

// ---- pack w2 into pair-interleaved buffer: w2pp[pp*256 + col*2 + {0,1}] ---
__global__ void k_init_w2pp(const float* __restrict__ w2, float* __restrict__ w2pp) {
  int total = NPAIR * HDIM;
  for (int i = blockIdx.x * blockDim.x + threadIdx.x; i < total;
       i += gridDim.x * blockDim.x) {
    int pp = i / HDIM, col = i - pp * HDIM;
    w2pp[(size_t)pp * (2 * HDIM) + col * 2 + 0] = w2row(w2, 2 * pp + 0, col);
    w2pp[(size_t)pp * (2 * HDIM) + col * 2 + 1] = w2row(w2, 2 * pp + 1, col);
  }
}

// ---- stage 1: rsum + scalar gather + 6->128 layer (one wave per node) -----
__global__ void k_stage1(const float* __restrict__ p,
                         const float* __restrict__ r_matrix,
                         const float* __restrict__ w1,
                         const float* __restrict__ b1,
                         const long long* __restrict__ idx,
                         float* __restrict__ rsum,
                         float* __restrict__ f1, int N) {
  int lane = threadIdx.x & 31;
  int n = blockIdx.x * (blockDim.x >> 5) + (threadIdx.x >> 5);
  if (n >= N) return;
  float r0 = 0.f, r1 = 0.f, r2 = 0.f, r3 = 0.f, r4 = 0.f, sp = 0.f;
  if (lane < KNB) {
    long long nb = idx[(size_t)n * (KNB + 1) + 1 + lane];
    sp = p[nb];
    const float* rp = r_matrix + ((size_t)n * KNB + lane) * 5;
    r0 = rp[0]; r1 = rp[1]; r2 = rp[2]; r3 = rp[3]; r4 = rp[4];
  }
  #pragma unroll
  for (int m = 16; m >= 1; m >>= 1) {
    sp += __shfl_xor(sp, m, 32);
    r0 += __shfl_xor(r0, m, 32);
    r1 += __shfl_xor(r1, m, 32);
    r2 += __shfl_xor(r2, m, 32);
    r3 += __shfl_xor(r3, m, 32);
    r4 += __shfl_xor(r4, m, 32);
  }
  if (lane == 0) {
    float* rs = rsum + (size_t)n * 5;
    rs[0] = r0; rs[1] = r1; rs[2] = r2; rs[3] = r3; rs[4] = r4;
  }
  float fd = 16.0f * p[n] - sp;
  #pragma unroll
  for (int j = 0; j < 4; j++) {
    int h = lane + 32 * j;
    float acc = 16.0f * b1[h]
              + r0 * w1[0 * HDIM + h] + r1 * w1[1 * HDIM + h]
              + r2 * w1[2 * HDIM + h] + r3 * w1[3 * HDIM + h]
              + r4 * w1[4 * HDIM + h] + fd * w1[5 * HDIM + h];
    f1[(size_t)n * HDIM + h] = fmaxf(acc, 0.0f);
  }
}

// ---- stage 2: float4 gather-sum into LDS, then WMMA f32 16x16x4 GEMM ------
__global__ void k_stage2(const float* __restrict__ f1,
                         const float* __restrict__ rsum,
                         const float* __restrict__ w2pp,
                         const float* __restrict__ b2,
                         const long long* __restrict__ idx,
                         float* __restrict__ f2, int N) {
  __shared__ float s2[16 * S2STRIDE];
  __shared__ int sidx[16 * KNB];
  int t = threadIdx.x;
  int base = blockIdx.x * 16;
  {
    int r = t >> 4, k = t & 15;
    int n = base + r; if (n >= N) n = N - 1;
    sidx[r * KNB + k] = (int)idx[(size_t)n * (KNB + 1) + 1 + k];
  }
  __syncthreads();

  // gather: thread handles 2 rows x 4 h-columns (float4, coalesced 512B/row)
  {
    int h0 = (t & 31) * 4;
    int rg = t >> 5;
    #pragma unroll
    for (int rr = 0; rr < 2; rr++) {
      int r = rg + rr * 8;
      int n = base + r; if (n >= N) n = N - 1;
      float gx = 0.f, gy = 0.f, gz = 0.f, gw = 0.f;
      #pragma unroll
      for (int k = 0; k < KNB; k++) {
        const float4 v = *(const float4*)(f1 + (size_t)sidx[r * KNB + k] * HDIM + h0);
        gx += v.x; gy += v.y; gz += v.z; gw += v.w;
      }
      const float4 fv = *(const float4*)(f1 + (size_t)n * HDIM + h0);
      float4 d;
      d.x = 16.0f * fv.x - gx; d.y = 16.0f * fv.y - gy;
      d.z = 16.0f * fv.z - gz; d.w = 16.0f * fv.w - gw;
      *(float4*)(&s2[r * S2STRIDE + DOFF + h0]) = d;
    }
  }
  if (t < 16) {
    int r = t; int n = base + r; if (n >= N) n = N - 1;
    const float* rs = rsum + (size_t)n * 5;
    s2[r * S2STRIDE + 0] = rs[0]; s2[r * S2STRIDE + 1] = rs[1];
    s2[r * S2STRIDE + 2] = rs[2]; s2[r * S2STRIDE + 3] = rs[3];
    s2[r * S2STRIDE + 4] = rs[4];
    s2[r * S2STRIDE + 5] = 0.0f; s2[r * S2STRIDE + 6] = 0.0f;
    s2[r * S2STRIDE + 7] = 0.0f;
  }
  __syncthreads();

  int wv = t >> 5, lane = t & 31;
  int row = lane & 15;       // M for A, N for B/C/D
  int hi = lane >> 4;        // selects K pair (A/B) and M+8 rows (C/D)
  int col = wv * 16 + row;   // output column in H
  v8f acc = {};
  for (int kc = 0; kc < KPAD / 4; kc++) {
    int kb = 4 * kc + hi * 2;
    int pp = 2 * kc + hi;
    v2f a = *(const v2f*)(&s2[row * S2STRIDE + kb]);                 // ds b64
    v2f b = *(const v2f*)(w2pp + (size_t)pp * (2 * HDIM) + col * 2); // global b64
    acc = __builtin_amdgcn_wmma_f32_16x16x4_f32(
        false, a, false, b, (short)0, acc, false, false);
  }
  float bias = 16.0f * b2[col];
  float* dst = f2 + (size_t)(base + hi * 8) * HDIM + col;
  if (base + 16 <= N) {
    #pragma unroll
    for (int r8 = 0; r8 < 8; r8++)
      dst[(size_t)r8 * HDIM] = fmaxf(acc[r8] + bias, 0.0f);
  } else {
    #pragma unroll
    for (int r8 = 0; r8 < 8; r8++)
      if (base + hi * 8 + r8 < N)
        dst[(size_t)r8 * HDIM] = fmaxf(acc[r8] + bias, 0.0f);
  }
}

// ---- stage 3: logits[n] = s3 . w3 + 16*b3 (one wave per node) -------------
__global__ void k_stage3(const float* __restrict__ f2,
                         const float* __restrict__ rsum,
                         const float* __restrict__ w3,
                         const float* __restrict__ b3,
                         const long long* __restrict__ idx,
                         float* __restrict__ logits, int N) {
  int lane = threadIdx.x & 31;
  int n = blockIdx.x * (blockDim.x >> 5) + (threadIdx.x >> 5);
  if (n >= N) return;
  const long long* ip = idx + (size_t)n * (KNB + 1) + 1;
  int h0 = lane * 4;
  float g0 = 0.f, g1 = 0.f, g2 = 0.f, g3 = 0.f;
  for (int k = 0; k < KNB; k++) {
    const float4 v = *(const float4*)(f2 + (size_t)ip[k] * HDIM + h0);
    g0 += v.x; g1 += v.y; g2 += v.z; g3 += v.w;
  }
  const float4 fv = *(const float4*)(f2 + (size_t)n * HDIM + h0);
  float acc = (16.0f * fv.x - g0) * w3[5 + h0]
            + (16.0f * fv.y - g1) * w3[6 + h0]
            + (16.0f * fv.z - g2) * w3[7 + h0]
            + (16.0f * fv.w - g3) * w3[8 + h0];
  if (lane == 0) {
    const float* rs = rsum + (size_t)n * 5;
    acc += rs[0] * w3[0] + rs[1] * w3[1] + rs[2] * w3[2]
         + rs[3] * w3[3] + rs[4] * w3[4] + 16.0f * b3[0];
  }
  #pragma unroll
  for (int m = 16; m >= 1; m >>= 1) acc += __shfl_xor(acc, m, 32);
  if (lane == 0) logits[n] = acc;
}

// ---- deterministic softmax over N ----------------------------------------
__global__ void k_max_partial(const float* __restrict__ logits,
                              float* __restrict__ part, int N) {
  __shared__ float sm[256];
  int t = threadIdx.x;
  float m = -INFINITY;
  for (int i = blockIdx.x * blockDim.x + t; i < N; i += gridDim.x * blockDim.x)
    m = fmaxf(m, logits[i]);
  sm[t] = m; __syncthreads();
  for (int s = 128; s >= 1; s >>= 1) {
    if (t < s) sm[t] = fmaxf(sm[t], sm[t + s]);
    __syncthreads();
  }
  if (t == 0) part[blockIdx.x] = sm[0];
}

__global__ void k_reduce_final(const float* __restrict__ part,
                               float* __restrict__ outv, int n, int is_max) {
  __shared__ float sm[256];
  int t = threadIdx.x;
  float v = is_max ? -INFINITY : 0.0f;
  if (t < n) v = part[t];
  sm[t] = v; __syncthreads();
  for (int s = 128; s >= 1; s >>= 1) {
    if (t < s) sm[t] = is_max ? fmaxf(sm[t], sm[t + s]) : (sm[t] + sm[t + s]);
    __syncthreads();
  }
  if (t == 0) outv[0] = sm[0];
}

__global__ void k_exp_partial(float* __restrict__ out,
                              const float* __restrict__ maxval,
                              float* __restrict__ part, int N) {
  __shared__ float sm[256];
  int t = threadIdx.x;
  float mv = maxval[0];
  float s = 0.0f;
  for (int i = blockIdx.x * blockDim.x + t; i < N; i += gridDim.x * blockDim.x) {
    float e = expf(out[i] - mv);
    out[i] = e; s += e;
  }
  sm[t] = s; __syncthreads();
  for (int k = 128; k >= 1; k >>= 1) {
    if (t < k) sm[t] += sm[t + k];
    __syncthreads();
  }
  if (t == 0) part[blockIdx.x] = sm[0];
}

__global__ void k_norm(float* __restrict__ out,
                       const float* __restrict__ sumval, int N) {
  float inv = 1.0f / sumval[0];
  for (int i = blockIdx.x * blockDim.x + threadIdx.x; i < N;
       i += gridDim.x * blockDim.x)
    out[i] *= inv;
}

extern "C" void kernel_launch(void* const* d_in, const int* in_sizes, int n_in,
                              void* d_out, int out_size, void* d_ws, size_t ws_size,
                              hipStream_t stream) {
  const float* p        = (const float*)d_in[0];
  const float* rmat     = (const float*)d_in[1];
  const float* w1       = (const float*)d_in[2];
  const float* b1       = (const float*)d_in[3];
  const float* w2       = (const float*)d_in[4];
  const float* b2       = (const float*)d_in[5];
  const float* w3       = (const float*)d_in[6];
  const float* b3       = (const float*)d_in[7];
  const long long* idx  = (const long long*)d_in[8];
  float* out = (float*)d_out;
  int N = in_sizes[0];

  float* ws   = (float*)d_ws;
  float* w2pp = ws;                         // NPAIR * 2*HDIM = KPAD*HDIM
  float* rsum = w2pp + KPAD * HDIM;         // N*5
  float* f1   = rsum + (size_t)N * 5;       // N*HDIM
  float* f2   = f1 + (size_t)N * HDIM;      // N*HDIM
  float* part = f2 + (size_t)N * HDIM;      // 256
  float* maxv = part + 256;                 // 1
  float* sumv = maxv + 1;                   // 1

  k_init_w2pp<<<68, 256, 0, stream>>>(w2, w2pp);

  int blocks_wave = (N + 7) / 8;            // 8 waves of 32 per 256-thread block
  k_stage1<<<blocks_wave, 256, 0, stream>>>(p, rmat, w1, b1, idx, rsum, f1, N);

  int tiles = (N + 15) / 16;
  k_stage2<<<tiles, 256, 0, stream>>>(f1, rsum, w2pp, b2, idx, f2, N);

  k_stage3<<<blocks_wave, 256, 0, stream>>>(f2, rsum, w3, b3, idx, out, N);

  k_max_partial<<<256, 256, 0, stream>>>(out, part, N);
  k_reduce_final<<<1, 256, 0, stream>>>(part, maxv, 256, 1);
  k_exp_partial<<<256, 256, 0, stream>>>(out, maxv, part, N);
  k_reduce_final<<<1, 256, 0, stream>>>(part, sumv, 256, 0);
  k_norm<<<256, 256, 0, stream>>>(out, sumv, N);
}